// MetaComprehensiveRegularization_36155034697795
// MI455X (gfx1250) — compile-verified
//
#include <hip/hip_runtime.h>
#include <hip/hip_bf16.h>
#include <math.h>

typedef float v2f __attribute__((ext_vector_type(2)));
typedef float v4f __attribute__((ext_vector_type(4)));
typedef float v8f __attribute__((ext_vector_type(8)));

#define N_ROWS   16384
#define D_DIM    512
#define N_VIEW   4
#define KSPLIT   4                       // waves cooperating on one 16-row tile (split along D)
#define TILES    (N_ROWS / 16)           // 1024
#define WG_SIZE  256
#define WAVES_PER_WG (WG_SIZE / 32)      // 8
#define WAVES_TOTAL  (TILES * KSPLIT)    // 4096
#define NUM_WGS      (WAVES_TOTAL / WAVES_PER_WG)  // 512
#define K_PER_WAVE   (D_DIM / KSPLIT)    // 128 floats of each row per wave
#define K_PER_HALF   (K_PER_WAVE / 2)    // 64 floats per lane-half
#define STEPS        (K_PER_HALF / 4)    // 16 float4 loads per stream

// D[m][n] += sum_k A[m][k]*B[k][n]; A and B fragments share the same per-lane
// pattern for the Gram use-case (lane m: k0,k1 of row m; lane m+16: k2,k3).
__device__ __forceinline__ v8f wmma_f32(v2f a, v2f b, v8f c) {
  return __builtin_amdgcn_wmma_f32_16x16x4_f32(
      /*neg_a=*/false, a, /*neg_b=*/false, b,
      /*c_mod=*/(short)0, c, /*reuse_a=*/false, /*reuse_b=*/false);
}

__device__ __forceinline__ float pick8(v8f v, int i) {
  float r = v[0];
  r = (i == 1) ? v[1] : r;
  r = (i == 2) ? v[2] : r;
  r = (i == 3) ? v[3] : r;
  r = (i == 4) ? v[4] : r;
  r = (i == 5) ? v[5] : r;
  r = (i == 6) ? v[6] : r;
  r = (i == 7) ? v[7] : r;
  return r;
}

extern "C" __global__ __launch_bounds__(WG_SIZE)
void mcr_gram_diag(const float* __restrict__ zc,
                   const float* __restrict__ zs,
                   float* __restrict__ pairs) {
  const int lane = threadIdx.x & 31;
  const int wv   = threadIdx.x >> 5;                 // wave index within WG
  const int gw   = blockIdx.x * WAVES_PER_WG + wv;   // global wave id
  const int tile = gw >> 2;                          // 16-row tile id (KSPLIT==4)
  const int ks   = gw & (KSPLIT - 1);                // which K-slice of the rows
  const int half = lane >> 4;                        // WMMA k-half owned by this lane
  const int row  = (tile << 4) + (lane & 15);

  const size_t off = (size_t)row * D_DIM
                   + (size_t)ks * K_PER_WAVE
                   + (size_t)half * K_PER_HALF;      // per-lane contiguous stream

  const v4f* __restrict__ pc = (const v4f*)(zc + off);
  const v4f* __restrict__ p0 = (const v4f*)(zs + 0 * (size_t)N_ROWS * D_DIM + off);
  const v4f* __restrict__ p1 = (const v4f*)(zs + 1 * (size_t)N_ROWS * D_DIM + off);
  const v4f* __restrict__ p2 = (const v4f*)(zs + 2 * (size_t)N_ROWS * D_DIM + off);
  const v4f* __restrict__ p3 = (const v4f*)(zs + 3 * (size_t)N_ROWS * D_DIM + off);

  v8f gcc = 0.0f;
  v8f gcs0 = 0.0f, gcs1 = 0.0f, gcs2 = 0.0f, gcs3 = 0.0f;
  v8f gss0 = 0.0f, gss1 = 0.0f, gss2 = 0.0f, gss3 = 0.0f;

#pragma unroll 4
  for (int i = 0; i < STEPS; ++i) {
    // Streaming, single-use data -> non-temporal b128 loads.
    v4f fc = __builtin_nontemporal_load(pc + i);
    v4f f0 = __builtin_nontemporal_load(p0 + i);
    v4f f1 = __builtin_nontemporal_load(p1 + i);
    v4f f2 = __builtin_nontemporal_load(p2 + i);
    v4f f3 = __builtin_nontemporal_load(p3 + i);

    v2f cL = {fc[0], fc[1]}, cH = {fc[2], fc[3]};
    gcc = wmma_f32(cL, cL, gcc);
    gcc = wmma_f32(cH, cH, gcc);

#define VIEW_STEP(fv, gcsv, gssv)                          \
    {                                                      \
      v2f sL = {fv[0], fv[1]}, sH = {fv[2], fv[3]};        \
      gcsv = wmma_f32(cL, sL, gcsv);                       \
      gcsv = wmma_f32(cH, sH, gcsv);                       \
      gssv = wmma_f32(sL, sL, gssv);                       \
      gssv = wmma_f32(sH, sH, gssv);                       \
    }
    VIEW_STEP(f0, gcs0, gss0)
    VIEW_STEP(f1, gcs1, gss1)
    VIEW_STEP(f2, gcs2, gss2)
    VIEW_STEP(f3, gcs3, gss3)
#undef VIEW_STEP
  }

  // --- extract Gram diagonals ---------------------------------------------
  // D layout (32-bit C/D 16x16): VGPR r holds M=r (lanes 0-15, N=lane) and
  // M=r+8 (lanes 16-31, N=lane-16). Diagonal m==n lives on:
  //   m=0..7  -> lane m,    d[m]
  //   m=8..15 -> lane m+16, d[m-8]
  __shared__ float part[WAVES_PER_WG][16][10];  // [wave][row-in-tile][9 sums], padded

  const bool active = (lane < 8) || (lane >= 24);
  const int  r      = (lane < 8) ? lane : (lane - 16);  // row-in-tile (valid if active)
  const int  di     = (lane < 8) ? lane : (lane - 24);  // index into v8f (valid if active)

  if (active) {
    part[wv][r][0] = pick8(gcc,  di);
    part[wv][r][1] = pick8(gcs0, di);
    part[wv][r][2] = pick8(gcs1, di);
    part[wv][r][3] = pick8(gcs2, di);
    part[wv][r][4] = pick8(gcs3, di);
    part[wv][r][5] = pick8(gss0, di);
    part[wv][r][6] = pick8(gss1, di);
    part[wv][r][7] = pick8(gss2, di);
    part[wv][r][8] = pick8(gss3, di);
  }
  __syncthreads();

  // --- one wave per tile combines the KSPLIT partial sums and normalizes ---
  if (ks == 0) {
    float c_p = 0.0f, s_p = 0.0f;
    if (active) {
      float cc = 0.0f, cs[N_VIEW] = {0, 0, 0, 0}, ss[N_VIEW] = {0, 0, 0, 0};
#pragma unroll
      for (int j = 0; j < KSPLIT; ++j) {
        cc    += part[wv + j][r][0];
        cs[0] += part[wv + j][r][1];
        cs[1] += part[wv + j][r][2];
        cs[2] += part[wv + j][r][3];
        cs[3] += part[wv + j][r][4];
        ss[0] += part[wv + j][r][5];
        ss[1] += part[wv + j][r][6];
        ss[2] += part[wv + j][r][7];
        ss[3] += part[wv + j][r][8];
      }
      const float nc = fmaxf(sqrtf(cc), 1e-12f);
      c_p = cc;
#pragma unroll
      for (int v = 0; v < N_VIEW; ++v) {
        const float ns = fmaxf(sqrtf(ss[v]), 1e-12f);
        s_p += cs[v] / (nc * ns);
      }
    }
    // full-wave reduction (inactive lanes contribute 0)
#pragma unroll
    for (int m = 16; m >= 1; m >>= 1) {
      c_p += __shfl_xor(c_p, m, 32);
      s_p += __shfl_xor(s_p, m, 32);
    }
    if (lane == 0) {
      pairs[2 * tile + 0] = c_p;
      pairs[2 * tile + 1] = s_p;
    }
  }
}

// Deterministic fixed-order final reduction over the 1024 tile pairs.
extern "C" __global__ __launch_bounds__(256)
void mcr_finalize(const float* __restrict__ pairs, float* __restrict__ out) {
  __shared__ float sc[256];
  __shared__ float ss[256];
  const int t = threadIdx.x;
  float c = 0.0f, s = 0.0f;
#pragma unroll
  for (int k = 0; k < TILES / 256; ++k) {
    const int idx = t + 256 * k;
    c += pairs[2 * idx + 0];
    s += pairs[2 * idx + 1];
  }
  sc[t] = c;
  ss[t] = s;
  __syncthreads();
#pragma unroll
  for (int stride = 128; stride >= 1; stride >>= 1) {
    if (t < stride) {
      sc[t] += sc[t + stride];
      ss[t] += ss[t + stride];
    }
    __syncthreads();
  }
  if (t == 0) {
    out[0] = -sc[0];  // loss_common
    out[1] = -ss[0];  // loss_special
  }
}

extern "C" void kernel_launch(void* const* d_in, const int* in_sizes, int n_in,
                              void* d_out, int out_size, void* d_ws, size_t ws_size,
                              hipStream_t stream) {
  const float* zc = (const float*)d_in[0];   // [16384, 512] f32
  const float* zs = (const float*)d_in[1];   // [4, 16384, 512] f32
  float* pairs = (float*)d_ws;               // 1024 * 2 floats = 8 KiB scratch
  float* out   = (float*)d_out;              // 2 floats

  mcr_gram_diag<<<NUM_WGS, WG_SIZE, 0, stream>>>(zc, zs, pairs);
  mcr_finalize<<<1, 256, 0, stream>>>(pairs, out);
}